// GridSquared2DParamsWithCoefficientsTorchPolynomial_62354335203767
// MI455X (gfx1250) — compile-verified
//
#include <hip/hip_runtime.h>

// ---------------------------------------------------------------------------
// out[b] = sum_{i,j} x^T M_{ij} x,  x = param[b,i,j,:] (P=64).
// Per cell: Y = X(32x64) @ M(64x64) via V_WMMA_F32_16X16X4_F32, fused dot.
// M stored K-pair interleaved so each B fragment is one ds_load_b64.
// Fragments preloaded -> single dscnt wait -> 16 back-to-back WMMAs (4 accs).
// Next cell's global data prefetched into registers during the WMMA phase.
// ---------------------------------------------------------------------------

typedef __attribute__((ext_vector_type(2))) float v2f;
typedef __attribute__((ext_vector_type(8))) float v8f;

#define NUM_WG       512
#define CELLS_PER_WG 8          // 512 * 8 = 4096 cells
#define PDIM         64
#define BDIM         32
#define NPAIR        2080
#define NCELL        4096
#define BSTRIDE      262144     // 64*64*64 floats between batches
#define XSTRIDE      68         // padded X row: bank = 4r mod 64, conflict-free
#define MPSTRIDE     160        // K-pair row: 128 data + 32 pad; halves in
                                // disjoint 32-bank windows for b64 B loads
#define DUMMY        140        // never-read pad slot (cols 128..159 of row 0)

__global__ __launch_bounds__(256)
void poly_quadform_kernel(const float* __restrict__ param,
                          const float* __restrict__ coeffs,
                          float* __restrict__ partial /* [NUM_WG*32] */) {
  __shared__ __align__(16) float Xs[BDIM * XSTRIDE];       // 8704 B
  __shared__ __align__(16) float Mp[(PDIM / 2) * MPSTRIDE]; // 20480 B
  __shared__ float acc[BDIM];

  const int tid  = threadIdx.x;
  const int lane = tid & 31;
  const int wave = tid >> 5;     // 0..7
  const int r    = lane & 15;
  const int half = lane >> 4;
  const int mt   = wave >> 2;    // batch tile 0..1
  const int nt   = wave & 3;     // col   tile 0..3

  if (tid < BDIM) acc[tid] = 0.0f;

  // ---- cell-invariant addressing ----------------------------------------
  // X staging (2 x float4 per thread)
  size_t xg_ofs[2];  int xl_ofs[2];
#pragma unroll
  for (int i = 0; i < 2; ++i) {
    const int f4 = tid + i * 256;            // 0..511 float4s
    const int b  = f4 >> 4;                  // 16 float4 per 64-float row
    const int p  = (f4 & 15) << 2;
    xg_ofs[i] = (size_t)b * BSTRIDE + p;
    xl_ofs[i] = b * XSTRIDE + p;
  }
  // coeff scatter: invert t = p*64 - p*(p-1)/2 + (q-p); write coeff at (p,q),
  // zero at mirrored (q,p); diagonal / invalid lanes hit the DUMMY pad slot.
  int addrU[9], addrL[9];
#pragma unroll
  for (int i = 0; i < 9; ++i) {
    const int t = tid + i * 256;
    if (t < NPAIR) {
      int p = (int)((129.0f - __builtin_sqrtf(16641.0f - 8.0f * (float)t)) * 0.5f);
      if (p < 0) p = 0;
      if (p > 63) p = 63;
      while ((((p + 1) * (129 - (p + 1))) >> 1) <= t) ++p;
      while (((p * (129 - p)) >> 1) > t) --p;
      const int q = p + (t - ((p * (129 - p)) >> 1));
      addrU[i] = (p >> 1) * MPSTRIDE + (q << 1) + (p & 1);
      addrL[i] = (p == q) ? DUMMY : (q >> 1) * MPSTRIDE + (p << 1) + (q & 1);
    } else {
      addrU[i] = DUMMY;
      addrL[i] = DUMMY + 1;
    }
  }
  // WMMA fragment addresses
  const int arow  = (mt * 16 + r) * XSTRIDE + 2 * half;  // A base (float idx)
  const int bcol2 = half * MPSTRIDE + ((nt * 16 + r) << 1); // B base (float idx)

  float facc[8];
#pragma unroll
  for (int v = 0; v < 8; ++v) facc[v] = 0.0f;

  const int cell0 = blockIdx.x * CELLS_PER_WG;

  // ---- prime the pipeline: prefetch first cell into registers ------------
  float4 xr[2];
  float  cr[9];
  {
    const float* xg = param  + (size_t)cell0 * PDIM;
    const float* cg = coeffs + (size_t)cell0 * NPAIR;
#pragma unroll
    for (int i = 0; i < 2; ++i) xr[i] = *(const float4*)(xg + xg_ofs[i]);
#pragma unroll
    for (int i = 0; i < 8; ++i) cr[i] = cg[tid + i * 256];
    cr[8] = (tid < NPAIR - 2048) ? cg[tid + 2048] : 0.0f;
  }

  for (int cc = 0; cc < CELLS_PER_WG; ++cc) {
    __syncthreads();  // previous cell's LDS reads complete (covers acc init)

    // ---- commit staged registers to LDS ----
#pragma unroll
    for (int i = 0; i < 2; ++i) *(float4*)&Xs[xl_ofs[i]] = xr[i];
#pragma unroll
    for (int i = 0; i < 9; ++i) {
      Mp[addrU[i]] = cr[i];
      Mp[addrL[i]] = 0.0f;
    }
    __syncthreads();

    // ---- prefetch next cell (overlaps with WMMA phase) ----
    {
      int nc = cell0 + cc + 1;
      if (nc >= NCELL) nc = NCELL - 1;      // clamped; result discarded
      const float* xg = param  + (size_t)nc * PDIM;
      const float* cg = coeffs + (size_t)nc * NPAIR;
#pragma unroll
      for (int i = 0; i < 2; ++i) xr[i] = *(const float4*)(xg + xg_ofs[i]);
#pragma unroll
      for (int i = 0; i < 8; ++i) cr[i] = cg[tid + i * 256];
      cr[8] = (tid < NPAIR - 2048) ? cg[tid + 2048] : 0.0f;
    }

    // ---- preload all fragments: one batch of DS loads, one wait ----
    v2f af[16], bf[16];
#pragma unroll
    for (int kb = 0; kb < 16; ++kb)
      af[kb] = *(const v2f*)&Xs[arow + 4 * kb];
#pragma unroll
    for (int kb = 0; kb < 16; ++kb)
      bf[kb] = *(const v2f*)&Mp[2 * kb * MPSTRIDE + bcol2];

    // ---- 16 WMMAs, 4 independent accumulator chains ----
    v8f c0 = {0.0f, 0.0f, 0.0f, 0.0f, 0.0f, 0.0f, 0.0f, 0.0f};
    v8f c1 = c0, c2 = c0, c3 = c0;
#pragma unroll
    for (int kb = 0; kb < 16; kb += 4) {
      c0 = __builtin_amdgcn_wmma_f32_16x16x4_f32(false, af[kb + 0], false, bf[kb + 0],
                                                 (short)0, c0, false, false);
      c1 = __builtin_amdgcn_wmma_f32_16x16x4_f32(false, af[kb + 1], false, bf[kb + 1],
                                                 (short)0, c1, false, false);
      c2 = __builtin_amdgcn_wmma_f32_16x16x4_f32(false, af[kb + 2], false, bf[kb + 2],
                                                 (short)0, c2, false, false);
      c3 = __builtin_amdgcn_wmma_f32_16x16x4_f32(false, af[kb + 3], false, bf[kb + 3],
                                                 (short)0, c3, false, false);
    }

    // ---- fused dot: facc += Y[b][q] * X[b][q]  (D VGPR v -> row v+8*half) ----
#pragma unroll
    for (int v = 0; v < 8; ++v) {
      const int brow = mt * 16 + half * 8 + v;
      facc[v] += ((c0[v] + c1[v]) + (c2[v] + c3[v])) *
                 Xs[brow * XSTRIDE + nt * 16 + r];
    }
  }

  // ---- reduce over q (16 lanes per half) with wave32 butterflies ----
#pragma unroll
  for (int v = 0; v < 8; ++v) {
    float val = facc[v];
#pragma unroll
    for (int off = 8; off >= 1; off >>= 1)
      val += __shfl_xor(val, off, 32);
    if (r == 0) atomicAdd(&acc[mt * 16 + half * 8 + v], val);
  }
  __syncthreads();
  if (tid < BDIM) partial[blockIdx.x * BDIM + tid] = acc[tid];
}

// ---- deterministic second stage: sum per-WG partials ----
__global__ void poly_reduce_kernel(const float* __restrict__ partial,
                                   float* __restrict__ out) {
  const int b = threadIdx.x;  // 0..31
  float s = 0.0f;
  for (int w = 0; w < NUM_WG; ++w) s += partial[w * BDIM + b];
  out[b] = s;
}

extern "C" void kernel_launch(void* const* d_in, const int* in_sizes, int n_in,
                              void* d_out, int out_size, void* d_ws, size_t ws_size,
                              hipStream_t stream) {
  const float* param  = (const float*)d_in[0];  // [32,64,64,64] f32
  const float* coeffs = (const float*)d_in[1];  // [64,64,2080]  f32
  float* partial = (float*)d_ws;                // NUM_WG*32 floats = 64 KB
  float* out = (float*)d_out;                   // [32] f32

  poly_quadform_kernel<<<NUM_WG, 256, 0, stream>>>(param, coeffs, partial);
  poly_reduce_kernel<<<1, 32, 0, stream>>>(partial, out);
}